// GCNConvDiffPool_42863773614895
// MI455X (gfx1250) — compile-verified
//
#include <hip/hip_runtime.h>
#include <math.h>

#define NN 10000
#define NE 640000
#define DF 256

typedef __attribute__((ext_vector_type(2))) float v2f;
typedef __attribute__((ext_vector_type(8))) float v8f;

// Guaranteed hardware FP atomic (global_atomic_add_f32 / ds_add_f32),
// avoiding a possible CAS-loop lowering of plain atomicAdd(float*).
static __device__ __forceinline__ float atomAddF(float* p, float v) {
#if defined(__HIP_DEVICE_COMPILE__)
    return unsafeAtomicAdd(p, v);
#else
    return atomicAdd(p, v);
#endif
}

// ---------------- init: deg = 1 (self-loop weight), xp1 = 0, reg = 0 -------
__global__ void k_init(float* __restrict__ deg, float* __restrict__ xp1,
                       float* __restrict__ reg) {
    int i = blockIdx.x * blockDim.x + threadIdx.x;
    if (i < NN) deg[i] = 1.0f;
    if (i < 144) xp1[i] = 0.0f;
    if (i == 0) reg[0] = 0.0f;
}

// ---------------- deg[dst] += ew ------------------------------------------
__global__ void k_deg(const int* __restrict__ src, const int* __restrict__ dst,
                      const float* __restrict__ ew, float* __restrict__ deg) {
    int e = blockIdx.x * blockDim.x + threadIdx.x;
    if (e < NE) atomAddF(&deg[dst[e]], ew[e]);
}

// ---------------- dinv = deg > 0 ? rsqrt(deg) : 0 -------------------------
__global__ void k_dinv(const float* __restrict__ deg, float* __restrict__ dinv) {
    int i = blockIdx.x * blockDim.x + threadIdx.x;
    if (i < NN) {
        float d = deg[i];
        dinv[i] = (d > 0.0f) ? __frsqrt_rn(d) : 0.0f;
    }
}

// ---------------- norm[e] = dinv[s]*ew*dinv[d] (shared by 3 passes) -------
__global__ void k_norm(const int* __restrict__ src, const int* __restrict__ dst,
                       const float* __restrict__ ew, const float* __restrict__ dinv,
                       float* __restrict__ norm) {
    int e = blockIdx.x * blockDim.x + threadIdx.x;
    if (e < NE) norm[e] = dinv[src[e]] * ew[e] * dinv[dst[e]];
}

// ---------------- y = x @ W1 via V_WMMA_F32_16X16X4_F32 -------------------
// One wave32 computes 16 rows. A(16x4) layout per ISA: v0 = K=2*half, v1 = K=2*half+1.
// B(4x16): only column N=0 carries W1 -> D column 0 holds the 16 dot products,
// in lane 0 (rows 0-7) and lane 16 (rows 8-15) across the 8 accumulator VGPRs.
__global__ void __launch_bounds__(32) k_gemv_wmma(const float* __restrict__ x,
                                                  const float* __restrict__ W1,
                                                  float* __restrict__ y) {
    const int lane = threadIdx.x;
    const int half = lane >> 4;    // 0: K+{0,1}, 1: K+{2,3}
    const int m    = lane & 15;    // row-in-tile (A) / column N (B)
    const size_t row = (size_t)blockIdx.x * 16 + m;
    const float* xr = x + row * DF + 2 * half;
    v8f c = {};
    v2f zero = {};
    for (int k = 0; k < DF; k += 4) {
        v2f a;
        a.x = xr[k];
        a.y = xr[k + 1];
        v2f wv;
        wv.x = W1[k + 2 * half];
        wv.y = W1[k + 2 * half + 1];
        v2f b = (m == 0) ? wv : zero;   // only output column N=0 is live
        c = __builtin_amdgcn_wmma_f32_16x16x4_f32(false, a, false, b,
                                                  (short)0, c, false, false);
    }
    if (m == 0) {
        float* yo = y + (size_t)blockIdx.x * 16 + half * 8;
#pragma unroll
        for (int r = 0; r < 8; ++r) yo[r] = c[r];
    }
}

// ------- out[i] = dinv[i]^2 * scale*in[i] + bias  (self-loop + bias) ------
__global__ void k_scatter_init(const float* __restrict__ tin,
                               const float* __restrict__ dinv,
                               const float* __restrict__ bias,
                               const float* __restrict__ scale, // nullptr -> 1
                               float* __restrict__ out) {
    int i = blockIdx.x * blockDim.x + threadIdx.x;
    if (i < NN) {
        float w = scale ? scale[0] : 1.0f;
        float dv = dinv[i];
        out[i] = dv * dv * w * tin[i] + bias[0];
    }
}

// ------- out[dst] += norm[e] * scale * in[src]  ---------------------------
__global__ void k_scatter_edge(const int* __restrict__ src, const int* __restrict__ dst,
                               const float* __restrict__ norm,
                               const float* __restrict__ tin,
                               const float* __restrict__ scale, // nullptr -> 1
                               float* __restrict__ out) {
    int e = blockIdx.x * blockDim.x + threadIdx.x;
    if (e < NE) {
        float w = scale ? scale[0] : 1.0f;
        atomAddF(&out[dst[e]], norm[e] * w * tin[src[e]]);
    }
}

// ------- pool 1 (fused): softmax(P1 row) -> entropy + S^T @ [h1,g2,g3] ----
__global__ void k_pool1(const float* __restrict__ P1,
                        const float* __restrict__ h1,
                        const float* __restrict__ g2,
                        const float* __restrict__ g3,
                        float* __restrict__ xp1, float* __restrict__ reg) {
    __shared__ float acc[144];
    __shared__ float entAcc;
    int tid = threadIdx.x;
    if (tid < 144) acc[tid] = 0.0f;
    if (tid == 0) entAcc = 0.0f;
    __syncthreads();

    int k = blockIdx.x * blockDim.x + tid;
    if (k < NN) {
        const float* p = P1 + (size_t)k * 48;
        float rowv[48];
        float mx = -INFINITY;
#pragma unroll
        for (int j = 0; j < 48; ++j) { rowv[j] = p[j]; mx = fmaxf(mx, rowv[j]); }
        float Z = 0.0f;
#pragma unroll
        for (int j = 0; j < 48; ++j) { rowv[j] = __expf(rowv[j] - mx); Z += rowv[j]; }
        float inv = 1.0f / Z;
        float hv0 = h1[k], hv1 = g2[k], hv2 = g3[k];
        float ent = 0.0f;
#pragma unroll
        for (int j = 0; j < 48; ++j) {
            float s = rowv[j] * inv;
            ent -= s * __logf(s + 1e-12f);
            atomAddF(&acc[j * 3 + 0], s * hv0);
            atomAddF(&acc[j * 3 + 1], s * hv1);
            atomAddF(&acc[j * 3 + 2], s * hv2);
        }
        atomAddF(&entAcc, ent);
    }
    __syncthreads();
    if (tid < 144) atomAddF(&xp1[tid], acc[tid]);
    if (tid == 0) atomAddF(reg, entAcc * (1.0f / (float)NN));
}

// ------- pools 2..4 + output (tiny: one thread) ---------------------------
__global__ void k_final(const float* __restrict__ xp1,
                        const float* __restrict__ P2,
                        const float* __restrict__ P3,
                        const float* __restrict__ P4,
                        const float* __restrict__ reg_in,
                        float* __restrict__ out) {
    if (threadIdx.x != 0 || blockIdx.x != 0) return;
    (void)P4; // softmax over 1 column == all-ones
    float reg = reg_in[0];

    // pool2: P2 is 48x12 ; xp2 = S2^T @ xp1
    float xp2[12 * 3];
    for (int t = 0; t < 36; ++t) xp2[t] = 0.0f;
    float ents = 0.0f;
    for (int r = 0; r < 48; ++r) {
        const float* p = P2 + r * 12;
        float e[12], mx = -INFINITY, Z = 0.0f;
        for (int c = 0; c < 12; ++c) mx = fmaxf(mx, p[c]);
        for (int c = 0; c < 12; ++c) { e[c] = __expf(p[c] - mx); Z += e[c]; }
        float inv = 1.0f / Z, ent = 0.0f;
        for (int c = 0; c < 12; ++c) {
            float s = e[c] * inv;
            ent -= s * __logf(s + 1e-12f);
            for (int n = 0; n < 3; ++n) xp2[c * 3 + n] += s * xp1[r * 3 + n];
        }
        ents += ent;
    }
    reg += ents / 48.0f;

    // pool3: P3 is 12x4 ; xp3 = S3^T @ xp2
    float xp3[4 * 3];
    for (int t = 0; t < 12; ++t) xp3[t] = 0.0f;
    ents = 0.0f;
    for (int r = 0; r < 12; ++r) {
        const float* p = P3 + r * 4;
        float e[4], mx = -INFINITY, Z = 0.0f;
        for (int c = 0; c < 4; ++c) mx = fmaxf(mx, p[c]);
        for (int c = 0; c < 4; ++c) { e[c] = __expf(p[c] - mx); Z += e[c]; }
        float inv = 1.0f / Z, ent = 0.0f;
        for (int c = 0; c < 4; ++c) {
            float s = e[c] * inv;
            ent -= s * __logf(s + 1e-12f);
            for (int n = 0; n < 3; ++n) xp3[c * 3 + n] += s * xp2[r * 3 + n];
        }
        ents += ent;
    }
    reg += ents / 12.0f;

    // pool4: S4 = ones(4,1); entropy per row = -log(1+1e-12); final = colsum(xp3)
    reg += -__logf(1.0f + 1e-12f);
    for (int n = 0; n < 3; ++n)
        out[n] = xp3[0 * 3 + n] + xp3[1 * 3 + n] + xp3[2 * 3 + n] + xp3[3 * 3 + n];
    out[3] = reg;
}

extern "C" void kernel_launch(void* const* d_in, const int* in_sizes, int n_in,
                              void* d_out, int out_size, void* d_ws, size_t ws_size,
                              hipStream_t stream) {
    (void)in_sizes; (void)n_in; (void)out_size; (void)ws_size;
    const float* x  = (const float*)d_in[0];
    const int*   ei = (const int*)d_in[1];   // (2,E) int32
    const float* ea = (const float*)d_in[2]; // (E,1)
    /* d_in[3] = adj : UNUSED by the reference */
    const float* W1 = (const float*)d_in[4];
    const float* b1 = (const float*)d_in[5];
    const float* W2 = (const float*)d_in[6];
    const float* b2 = (const float*)d_in[7];
    const float* P1 = (const float*)d_in[8];
    const float* P2 = (const float*)d_in[9];
    const float* P3 = (const float*)d_in[10];
    const float* P4 = (const float*)d_in[11];
    float* out = (float*)d_out;

    float* ws   = (float*)d_ws;
    float* deg  = ws;
    float* dinv = ws + NN;
    float* y    = ws + 2 * NN;
    float* h1   = ws + 3 * NN;
    float* g2   = ws + 4 * NN;
    float* g3   = ws + 5 * NN;
    float* norm = ws + 6 * NN;       // NE floats
    float* xp1  = norm + NE;         // 144 floats
    float* reg  = xp1 + 144;         // 1 float

    const int* src = ei;
    const int* dst = ei + NE;

    const int TB = 256;
    const int gN = (NN + TB - 1) / TB;
    const int gE = (NE + TB - 1) / TB;

    k_init<<<gN, TB, 0, stream>>>(deg, xp1, reg);
    k_deg<<<gE, TB, 0, stream>>>(src, dst, ea, deg);
    k_dinv<<<gN, TB, 0, stream>>>(deg, dinv);
    k_norm<<<gE, TB, 0, stream>>>(src, dst, ea, dinv, norm);

    k_gemv_wmma<<<NN / 16, 32, 0, stream>>>(x, W1, y);

    // pass 1: h1 from y (scale=1, bias=b1)
    k_scatter_init<<<gN, TB, 0, stream>>>(y, dinv, b1, nullptr, h1);
    k_scatter_edge<<<gE, TB, 0, stream>>>(src, dst, norm, y, nullptr, h1);
    // pass 2: g2 from h1 (scale=W2, bias=b2)
    k_scatter_init<<<gN, TB, 0, stream>>>(h1, dinv, b2, W2, g2);
    k_scatter_edge<<<gE, TB, 0, stream>>>(src, dst, norm, h1, W2, g2);
    // pass 3: g3 from g2 (scale=W2, bias=b2)
    k_scatter_init<<<gN, TB, 0, stream>>>(g2, dinv, b2, W2, g3);
    k_scatter_edge<<<gE, TB, 0, stream>>>(src, dst, norm, g2, W2, g3);

    k_pool1<<<gN, TB, 0, stream>>>(P1, h1, g2, g3, xp1, reg);
    k_final<<<1, 32, 0, stream>>>(xp1, P2, P3, P4, reg, out);
}